// AllToAllDispatcher_86242943303726
// MI455X (gfx1250) — compile-verified
//
#include <hip/hip_runtime.h>
#include <hip/hip_bf16.h>
#include <stdint.h>

// Problem constants (from reference)
#define NE      64      // experts
#define TOPK    2
#define NTOK    32768
#define FLAT    (NTOK * TOPK)   // 65536 flattened routing entries
#define DM      1024            // d_model
#define CHUNK   256             // keys per chunk (block)
#define NCHUNK  (FLAT / CHUNK)  // 256 chunks

// Workspace layout (as int*):
//   ws[0 .. 64)                      : per-expert totals (atomic)
//   ws[64 .. 64+NCHUNK*NE)           : per-chunk histograms  hist[chunk][expert]
//   ws[WS_COFF .. +NCHUNK*NE)        : per-chunk/per-expert exclusive offsets
//   ws[WS_PSRC .. +FLAT)             : perm_src[row] = source token index
#define WS_HIST   64
#define WS_COFF   (64 + NCHUNK * NE)
#define WS_PSRC   (64 + 2 * NCHUNK * NE)

__global__ void k_init(int* __restrict__ ws) {
    if (threadIdx.x < NE) ws[threadIdx.x] = 0;
}

// Per-chunk histogram + global per-expert totals.
__global__ void k_hist(const int* __restrict__ routing, int* __restrict__ ws) {
    __shared__ int h[NE];
    const int tid = threadIdx.x;
    const int c   = blockIdx.x;
    if (tid < NE) h[tid] = 0;
    __syncthreads();
    const int key = routing[c * CHUNK + tid];
    atomicAdd(&h[key], 1);
    __syncthreads();
    if (tid < NE) {
        ws[WS_HIST + c * NE + tid] = h[tid];
        atomicAdd(&ws[tid], h[tid]);
    }
}

// One block per expert (grid = 64). 256-wide parallel exclusive scan of that
// expert's per-chunk histogram column; adds the expert base offset.
__global__ void __launch_bounds__(CHUNK) k_scan(int* __restrict__ ws,
                                                float* __restrict__ out_counts) {
    __shared__ int s[CHUNK];
    __shared__ int tot[NE];
    const int e   = blockIdx.x;
    const int tid = threadIdx.x;
    if (tid < NE) tot[tid] = ws[tid];
    const int own = ws[WS_HIST + tid * NE + e];
    s[tid] = own;
    __syncthreads();

    // expert base = exclusive sum of totals below e (redundant per-thread, LDS broadcast)
    int base = 0;
    for (int k = 0; k < e; ++k) base += tot[k];

    // Hillis-Steele inclusive scan over 256 chunks
    #pragma unroll
    for (int off = 1; off < CHUNK; off <<= 1) {
        int t = (tid >= off) ? s[tid - off] : 0;
        __syncthreads();
        s[tid] += t;
        __syncthreads();
    }
    const int excl = s[tid] - own;
    ws[WS_COFF + tid * NE + e] = base + excl;
    if (tid == 0) out_counts[e] = (float)tot[e];
}

// Stable destination of every flattened routing entry.
__global__ void k_rank(const int* __restrict__ routing, int* __restrict__ ws,
                       float* __restrict__ out_inv, float* __restrict__ out_esi) {
    __shared__ int keys[CHUNK];
    const int tid = threadIdx.x;
    const int c   = blockIdx.x;
    const int ig  = c * CHUNK + tid;
    const int key = routing[ig];
    keys[tid] = key;
    __syncthreads();
    int rank = 0;
    for (int j = 0; j < tid; ++j) rank += (keys[j] == key) ? 1 : 0;
    const int* coff = ws + WS_COFF;
    const int dst = coff[c * NE + key] + rank;
    out_inv[dst] = (float)ig;          // inverse_permute_mapping = argsort(flat)
    out_esi[ig]  = (float)ig;          // stable argsort of sorted keys == identity
    int* perm_src = ws + WS_PSRC;
    perm_src[dst] = ig >> 1;           // token index (TOPK == 2)
}

// Bandwidth kernel: 1 wave per 4KB output row, staged through LDS with the
// CDNA5 async global<->LDS path (ASYNCcnt). Loads stay RT (128MB hidden fits
// in 192MB L2 and is read ~2x); stores are NT (write-once, never re-read) so
// they don't evict the resident input.
__global__ void __launch_bounds__(256) k_gather(const float* __restrict__ hidden,
                                                const int* __restrict__ perm_src,
                                                float* __restrict__ out) {
    __shared__ float smem[8 * DM];       // 8 waves x 4KB staging
    const int wave = threadIdx.x >> 5;
    const int lane = threadIdx.x & 31;
    const int row  = blockIdx.x * 8 + wave;
    const int src_row = perm_src[row];

    // Low 32 bits of the generic shared address == LDS byte offset.
    const unsigned lbase = (unsigned)(uintptr_t)(&smem[wave * DM]) + (unsigned)(lane * 16);
    const unsigned gsrc  = (unsigned)src_row * (DM * 4u) + (unsigned)(lane * 16);
    const unsigned gdst  = (unsigned)row     * (DM * 4u) + (unsigned)(lane * 16);

    // 8 x (32 lanes x 16B) = 4KB row: global -> LDS (async)
#pragma unroll
    for (int k = 0; k < 8; ++k) {
        asm volatile("global_load_async_to_lds_b128 %0, %1, %2"
                     :: "v"(lbase + (unsigned)(k * 512)),
                        "v"(gsrc + (unsigned)(k * 512)),
                        "s"(hidden)
                     : "memory");
    }
    asm volatile("s_wait_asynccnt 0" ::: "memory");
    // LDS -> global (async, non-temporal)
#pragma unroll
    for (int k = 0; k < 8; ++k) {
        asm volatile("global_store_async_from_lds_b128 %0, %1, %2 th:TH_STORE_NT"
                     :: "v"(gdst + (unsigned)(k * 512)),
                        "v"(lbase + (unsigned)(k * 512)),
                        "s"(out)
                     : "memory");
    }
    // S_ENDPGM performs an implicit S_WAIT_IDLE (covers ASYNCcnt) before retiring.
}

extern "C" void kernel_launch(void* const* d_in, const int* in_sizes, int n_in,
                              void* d_out, int out_size, void* d_ws, size_t ws_size,
                              hipStream_t stream) {
    const float* hidden  = (const float*)d_in[0];   // [32768, 1024] f32
    const int*   routing = (const int*)d_in[1];     // [32768, 2] int

    float* out      = (float*)d_out;
    float* out_inv  = out + (size_t)FLAT * DM;      // inverse_permute_mapping
    float* out_esi  = out_inv + FLAT;               // expert_sort_indices
    float* out_cnt  = out_esi + FLAT;               // per-expert counts

    int* ws = (int*)d_ws;
    const int* perm_src = ws + WS_PSRC;

    k_init<<<1, 64, 0, stream>>>(ws);
    k_hist<<<NCHUNK, CHUNK, 0, stream>>>(routing, ws);
    k_scan<<<NE, CHUNK, 0, stream>>>(ws, out_cnt);
    k_rank<<<NCHUNK, CHUNK, 0, stream>>>(routing, ws, out_inv, out_esi);
    k_gather<<<FLAT / 8, 256, 0, stream>>>(hidden, perm_src, out);
}